// IntegratedConsistencyADPIT_36996848287751
// MI455X (gfx1250) — compile-verified
//
#include <hip/hip_runtime.h>

// ADPIT loss for MI455X (gfx1250, wave32). Memory-bound: ~155 MB streamed once
// -> ~6.6 us at 23.3 TB/s. Round-3 changes vs round 2 (which spilled ~35 VMEM
// worth of scratch traffic at ~200 live VGPRs):
//  * v2f / global_load_b64 (6 threads per (b,t)): 256 B per wave-load, half the
//    register liveness of the v4f version.
//  * Slot-streaming: per track-slot, load 3 noisy + 3 clean pairs, build only
//    the needed padded vectors g[x][y] = A0+B_x+C_y on the fly, and fold the
//    squared distances straight into 11 candidate accumulators (all 13
//    permutations collapse to these; combined[0]==combined[1]==combined[7]).
//    Peak liveness ~90 VGPRs -> no spills, high occupancy for latency hiding.
//  * Exact fp32 wave sum via V_WMMA_F32_16X16X4_F32 ones-trick + shfl_xor(16).
//  * Deterministic two-pass reduction through d_ws.

typedef __attribute__((ext_vector_type(2))) float v2f;
typedef __attribute__((ext_vector_type(8))) float v8f;

#define BDIM   128
#define TDIM   600
#define CDIM   12
#define BT_TOTAL (BDIM * TDIM)            // 76800
#define NELEM    (BT_TOTAL * CDIM)        // 921600 elements (b,t,c)
#define PAIRS    (CDIM / 2)               // 6 channel-pairs per (b,t)
#define NTHREADS (BT_TOTAL * PAIRS)       // 460800
#define BLOCK    256
#define NBLOCK   (NTHREADS / BLOCK)       // 1800 (exact)

// Exact fp32 sum of the 32 lane values of a wave using one WMMA.
// A(16x4): a0 = v (lanes 0-15 -> K=0; lanes 16-31 -> K=2), a1 = 0.
// B(4x16) = ones  =>  D[m][n] = v_m + v_{m+16} = s_m.
// Lane l holds s_{r + 8*(l>>4)} in acc VGPR r; in-lane sum of the 8 acc
// components gives S_lo / S_hi; shfl_xor(16) combines halves.
// EXEC must be all ones at the call site.
__device__ __forceinline__ float wave_sum_wmma(float v) {
  v2f a; a[0] = v;    a[1] = 0.0f;
  v2f b; b[0] = 1.0f; b[1] = 1.0f;
  v8f c = {};
  v8f d = __builtin_amdgcn_wmma_f32_16x16x4_f32(
      false, a, false, b, (short)0, c, false, false);
  float s = d[0] + d[1] + d[2] + d[3] + d[4] + d[5] + d[6] + d[7];
  s += __shfl_xor(s, 16, 32);
  return s;
}

__device__ __forceinline__ float block_sum(float v, float* wsum) {
  const float ws = wave_sum_wmma(v);
  const int lane = threadIdx.x & 31;
  const int wave = threadIdx.x >> 5;
  if (lane == 0) wsum[wave] = ws;
  __syncthreads();
  float s = 0.0f;
  if (threadIdx.x == 0) {
#pragma unroll
    for (int w = 0; w < BLOCK / 32; ++w) s += wsum[w];
  }
  return s;  // valid in thread 0 only
}

__device__ __forceinline__ v2f vmin2(v2f a, v2f b) {
  v2f r;
  r[0] = fminf(a[0], b[0]);
  r[1] = fminf(a[1], b[1]);
  return r;
}

__global__ void adpit_partial_kernel(const float* __restrict__ noisy,
                                     const float* __restrict__ clean,
                                     const float* __restrict__ labels,
                                     const float* __restrict__ weights,
                                     float* __restrict__ partial) {
  const int idx = blockIdx.x * BLOCK + threadIdx.x;  // < NTHREADS exactly
  const int g   = idx % PAIRS;                       // channel pair
  const int bt  = idx / PAIRS;
  const int bb  = bt / TDIM;

  // v2f-unit bases: noisy/clean tile = 54 v2f per (b,t); labels = 144 v2f.
  const v2f* np = (const v2f*)noisy  + (size_t)bt * 54 + g;
  const v2f* cp = (const v2f*)clean  + (size_t)bt * 54 + g;
  const v2f* lp = (const v2f*)labels + (size_t)bt * 144 + g;

  // Speculative streaming prefetch (global_prefetch_b8); OOB silently dropped.
  __builtin_prefetch(np + 128 * 54, 0, 0);
  __builtin_prefetch(cp + 128 * 54, 0, 0);
  __builtin_prefetch(lp + 128 * 144, 0, 0);

  // Activity-weighted targets: 0=A0 1=B0 2=B1 3=C0 4=C1 5=C2 ; t[k][i] v2f.
  v2f t[6][3];
#pragma unroll
  for (int k = 0; k < 6; ++k) {
    const v2f act = lp[(k * 4) * 6];
#pragma unroll
    for (int i = 0; i < 3; ++i) t[k][i] = act * lp[(k * 4 + 1 + i) * 6];
  }

  // Which E[s][x][y] are required (12 of 18):
  //  s=0: x0:{y0,y1,y2}, x1:{y0}
  //  s=1: x0:{y0,y1,y2}, x1:{y1}
  //  s=2: x1:{y0,y1,y2}, x0:{y2}
  constexpr bool need[3][2][3] = {
      {{true, true, true}, {true, false, false}},
      {{true, true, true}, {false, true, false}},
      {{false, false, true}, {true, true, true}}};
  // C-group candidates: slot pattern x=(0,0,1), y = permutation of (0,1,2).
  constexpr int yperm[6][3] = {{0,1,2},{0,2,1},{1,0,2},{1,2,0},{2,0,1},{2,1,0}};
  // B-group candidates: y=s, x-patterns (p2..p6).
  constexpr int xpat[5][3] = {{0,1,0},{0,1,1},{1,0,0},{1,0,1},{1,1,0}};

  v2f candC[6] = {};
  v2f candB[5] = {};
  v2f cons = {};

  // Stream the 3 track-slots; keep only this slot's data + accumulators live.
#pragma unroll
  for (int s = 0; s < 3; ++s) {
    v2f nzs[3];
#pragma unroll
    for (int i = 0; i < 3; ++i) {
      nzs[i] = np[(3 * s + i) * 6];
      const v2f dd = nzs[i] - cp[(3 * s + i) * 6];
      cons += dd * dd;
    }

    v2f E[2][3];  // only 'need'ed entries are written/read (full unroll)
#pragma unroll
    for (int x = 0; x < 2; ++x) {
#pragma unroll
      for (int y = 0; y < 3; ++y) {
        if (need[s][x][y]) {
          v2f acc = {};
#pragma unroll
          for (int i = 0; i < 3; ++i) {
            const v2f gvi = t[0][i] + t[1 + x][i] + t[3 + y][i];
            const v2f dd = nzs[i] - gvi;
            acc += dd * dd;
          }
          E[x][y] = acc;
        }
      }
    }

#pragma unroll
    for (int j = 0; j < 6; ++j)
      candC[j] += E[(s == 2) ? 1 : 0][yperm[j][s]];
#pragma unroll
    for (int j = 0; j < 5; ++j)
      candB[j] += E[xpat[j][s]][s];
  }

  // Min over the 11 distinct candidates (== min over all 13 permutations).
  v2f best = candC[0];
#pragma unroll
  for (int j = 1; j < 6; ++j) best = vmin2(best, candC[j]);
#pragma unroll
  for (int j = 0; j < 5; ++j) best = vmin2(best, candB[j]);

  const float w = weights[bb];
  const v2f fin = best * (1.0f / 9.0f) + w * (cons * (1.0f / 9.0f));
  const float val = fin[0] + fin[1];

  __shared__ float wsum[BLOCK / 32];
  const float s = block_sum(val, wsum);
  if (threadIdx.x == 0) partial[blockIdx.x] = s;
}

__global__ void adpit_finalize_kernel(const float* __restrict__ partial,
                                      float* __restrict__ out) {
  float acc = 0.0f;
  for (int j = threadIdx.x; j < NBLOCK; j += BLOCK) acc += partial[j];
  __shared__ float wsum[BLOCK / 32];
  const float s = block_sum(acc, wsum);  // converged -> EXEC all ones
  if (threadIdx.x == 0) out[0] = s * (1.0f / (float)NELEM);
}

extern "C" void kernel_launch(void* const* d_in, const int* in_sizes, int n_in,
                              void* d_out, int out_size, void* d_ws, size_t ws_size,
                              hipStream_t stream) {
  const float* noisy   = (const float*)d_in[0];
  const float* clean   = (const float*)d_in[1];
  const float* labels  = (const float*)d_in[2];
  const float* weights = (const float*)d_in[3];
  float* partial = (float*)d_ws;  // NBLOCK floats = 7.2 KB scratch

  adpit_partial_kernel<<<NBLOCK, BLOCK, 0, stream>>>(noisy, clean, labels,
                                                     weights, partial);
  adpit_finalize_kernel<<<1, BLOCK, 0, stream>>>(partial, (float*)d_out);
}